// GatedGraphConv_66786741453226
// MI455X (gfx1250) — compile-verified
//
#include <hip/hip_runtime.h>

typedef float v2f __attribute__((ext_vector_type(2)));
typedef float v8f __attribute__((ext_vector_type(8)));

#define IN_F  64
#define OUT_F 32
#define NB    2
#define NN    2048
#define BN    (NB * NN)   // 4096 total node rows

// ---------------------------------------------------------------------------
// Kernel 1: fused projection  x @ {A_w, B_w, V, U}  via V_WMMA_F32_16X16X4_F32
// One wave computes a 16-row tile of all four [16 x 32] outputs.
//   A frag (16x4 f32, ISA 7.12.2): lanes 0-15 -> K0(K1), lanes 16-31 -> K2(K3)
//   B frag (4x16 f32): same K split across lane halves, N = lane&15
//   D (16x16 f32): VGPR v -> row v (lanes 0-15) / row v+8 (lanes 16-31)
// ---------------------------------------------------------------------------
__global__ __launch_bounds__(256) void ggc_proj_wmma(
    const float* __restrict__ x,
    const float* __restrict__ U,  const float* __restrict__ V,
    const float* __restrict__ Aw, const float* __restrict__ Bw,
    float* __restrict__ Ah, float* __restrict__ Bh, float* __restrict__ Vh,
    float* __restrict__ out)
{
    const int lane = threadIdx.x & 31;
    const int wid  = threadIdx.x >> 5;
    const int tile = blockIdx.x * 8 + wid;   // one 16-row tile per wave
    const int row0 = tile * 16;
    const int m    = lane & 15;
    const int half = lane >> 4;

    const float* Ws[4] = {Aw, Bw, V, U};

    v8f acc[8];
    #pragma unroll
    for (int i = 0; i < 8; ++i) acc[i] = {};

    // x row for this lane's M, offset by this lane-half's K sub-slot
    const float* xr = x + (size_t)(row0 + m) * IN_F + 2 * half;

    #pragma unroll
    for (int kk = 0; kk < IN_F / 4; ++kk) {
        const int kbase = kk * 4;
        v2f a;
        a.x = xr[kbase + 0];     // x[row0+m][kbase + 2*half + 0]
        a.y = xr[kbase + 1];     // x[row0+m][kbase + 2*half + 1]
        const int kb = kbase + 2 * half;
        #pragma unroll
        for (int w = 0; w < 4; ++w) {
            #pragma unroll
            for (int nt = 0; nt < 2; ++nt) {
                v2f b;
                b.x = Ws[w][(size_t)(kb + 0) * OUT_F + nt * 16 + m];
                b.y = Ws[w][(size_t)(kb + 1) * OUT_F + nt * 16 + m];
                // 8 args: (neg_a, A, neg_b, B, c_mod, C, reuse_a, reuse_b)
                acc[w * 2 + nt] = __builtin_amdgcn_wmma_f32_16x16x4_f32(
                    false, a, false, b, (short)0, acc[w * 2 + nt], false, false);
            }
        }
    }

    float* Ds[4] = {Ah, Bh, Vh, out};   // U-result goes straight to d_out
    #pragma unroll
    for (int w = 0; w < 4; ++w) {
        #pragma unroll
        for (int nt = 0; nt < 2; ++nt) {
            #pragma unroll
            for (int v = 0; v < 8; ++v) {
                const int r = row0 + v + 8 * half;
                Ds[w][(size_t)r * OUT_F + nt * 16 + m] = acc[w * 2 + nt][v];
            }
        }
    }
}

// ---------------------------------------------------------------------------
// Kernel 2: gated aggregation.
//   out[row, o] += sum_j sigmoid(Ah[row,o] + Bh[j,o]) * adj[row,j] * Vh[j,o]
// One wave per node row (lane == output channel). adj is the only HBM-heavy
// stream (33.5 MB, read once, coalesced 32 j's at a time, broadcast by shfl).
// Bh/Vh j-tiles are staged in LDS and shared by the 8 waves of the block
// (all 8 rows of a block belong to the same batch since 2048 % 8 == 0).
// ---------------------------------------------------------------------------
__global__ __launch_bounds__(256) void ggc_gate_agg(
    const float* __restrict__ adj,
    const float* __restrict__ Ah, const float* __restrict__ Bh,
    const float* __restrict__ Vh, float* __restrict__ out)
{
    __shared__ float sB[32 * OUT_F];
    __shared__ float sV[32 * OUT_F];

    const int lane = threadIdx.x & 31;
    const int wid  = threadIdx.x >> 5;
    const int row  = blockIdx.x * 8 + wid;    // global node row in [0, BN)
    const int b    = row >> 11;               // batch index (row / 2048)

    const float* Bhb    = Bh  + (size_t)b * NN * OUT_F;
    const float* Vhb    = Vh  + (size_t)b * NN * OUT_F;
    const float* adjRow = adj + (size_t)row * NN;
    const float  ah     = Ah[(size_t)row * OUT_F + lane];

    float acc = 0.0f;
    for (int jbase = 0; jbase < NN; jbase += 32) {
        __syncthreads();                       // previous tile fully consumed
        #pragma unroll
        for (int k = 0; k < (32 * OUT_F) / 256; ++k) {
            const int idx = threadIdx.x + k * 256;
            sB[idx] = Bhb[(size_t)jbase * OUT_F + idx];
            sV[idx] = Vhb[(size_t)jbase * OUT_F + idx];
        }
        __syncthreads();

        const float av = adjRow[jbase + lane]; // coalesced adj read, once
        #pragma unroll
        for (int jj = 0; jj < 32; ++jj) {
            const float adjv = __shfl(av, jj, 32);
            const float bh   = sB[jj * OUT_F + lane];
            const float vh   = sV[jj * OUT_F + lane];
            // sigmoid via v_exp_f32 + v_rcp_f32
            const float s = __builtin_amdgcn_rcpf(1.0f + __expf(-(ah + bh)));
            acc = fmaf(adjv * s, vh, acc);
        }
    }

    out[(size_t)row * OUT_F + lane] += acc;    // adds onto x@U from kernel 1
}

extern "C" void kernel_launch(void* const* d_in, const int* in_sizes, int n_in,
                              void* d_out, int out_size, void* d_ws, size_t ws_size,
                              hipStream_t stream) {
    const float* adj = (const float*)d_in[0];  // [B,N,N]
    const float* x   = (const float*)d_in[1];  // [B,N,64]
    const float* U   = (const float*)d_in[2];  // [64,32]
    const float* V   = (const float*)d_in[3];  // [64,32]
    const float* Aw  = (const float*)d_in[4];  // [64,32]
    const float* Bw  = (const float*)d_in[5];  // [64,32]
    float* out = (float*)d_out;                // [B,N,32]

    float* ws = (float*)d_ws;
    float* Ah = ws;                                  // 4096*32 f32 = 512 KB
    float* Bh = ws + (size_t)BN * OUT_F;
    float* Vh = ws + (size_t)2 * BN * OUT_F;

    // 4096 rows / (16 rows per wave * 8 waves per block) = 32 blocks
    ggc_proj_wmma<<<BN / (16 * 8), 256, 0, stream>>>(x, U, V, Aw, Bw,
                                                     Ah, Bh, Vh, out);
    // 4096 rows / 8 rows per block = 512 blocks
    ggc_gate_agg<<<BN / 8, 256, 0, stream>>>(adj, Ah, Bh, Vh, out);
}